// MultiHeadAttentionQuantum_65481071399182
// MI455X (gfx1250) — compile-verified
//
#include <hip/hip_runtime.h>
#include <hip/hip_bf16.h>
#include <math.h>

// Problem constants (match reference)
#define B_ 8
#define S_ 4096
#define E_ 768
#define H_ 12
#define NW_ 8
#define DK_ 64
#define NTOK (B_ * S_)          // 32768 tokens
#define TOKS_PER_BLOCK 128      // 8 waves x 16 tokens

typedef __attribute__((ext_vector_type(2))) float v2f;
typedef __attribute__((ext_vector_type(8))) float v8f;

// ---------------------------------------------------------------------------
// Kernel 1: fold theta/W_proj/W_out into an effective rank-16 weight matrix
//   M[r][e]   r in [0,16): r<8 -> cos branch, r>=8 -> -sin branch
//   b_eff[e]  = b_proj_flat @ W_out + b_out
// grid = 51 blocks * 256 thr covers 17*768 outputs (rows 0..15 = M, 16 = bias)
// ---------------------------------------------------------------------------
__global__ __launch_bounds__(256) void qmha_prep(
    const float* __restrict__ theta,  // [12,8]
    const float* __restrict__ Wp,     // [12,8,64]
    const float* __restrict__ bp,     // [12,64]
    const float* __restrict__ Wo,     // [768,768] (row = input j, col = output e)
    const float* __restrict__ bo,     // [768]
    float* __restrict__ M,            // [16,768]
    float* __restrict__ beff)         // [768]
{
    int idx = blockIdx.x * blockDim.x + threadIdx.x;
    int r = idx / E_;
    int e = idx - r * E_;

    if (r < 16) {
        int w = r & 7;
        bool isCos = (r < 8);
        float acc = 0.f;
        for (int h = 0; h < H_; ++h) {
            float th = theta[h * NW_ + w];
            float cs = isCos ? cosf(th) : -sinf(th);
            const float* wp = Wp + (h * NW_ + w) * DK_;
            const float* wo = Wo + (size_t)(h * DK_) * E_ + e;
            #pragma unroll 8
            for (int d = 0; d < DK_; ++d)
                acc = fmaf(cs * wp[d], wo[(size_t)d * E_], acc);
        }
        M[r * E_ + e] = acc;
    } else if (r == 16) {
        float acc = bo[e];
        for (int j = 0; j < E_; ++j)
            acc = fmaf(bp[j], Wo[(size_t)j * E_ + e], acc);
        beff[e] = acc;
    }
}

// ---------------------------------------------------------------------------
// Kernel 2: out[token,:] = [cos(x0..7), sin(x0..7)] @ M + b_eff
// 256 threads = 8 waves; wave w owns tokens [base + 16w, base + 16w + 16).
// fp32 WMMA 16x16x4, K=16 via 4 chained WMMAs; 48 column tiles of 16.
// ---------------------------------------------------------------------------
__global__ __launch_bounds__(256) void qmha_main(
    const float* __restrict__ x,     // [NTOK, 768] (only first 8 cols used)
    const float* __restrict__ M,     // [16, 768]
    const float* __restrict__ beff,  // [768]
    float* __restrict__ out)         // [NTOK, 768]
{
    __shared__ float fLDS[TOKS_PER_BLOCK][16];   // cos|sin features, 8 KB
    __shared__ float mLDS[16][E_];               // rank-16 weights, 48 KB
    __shared__ float bLDS[E_];                   // bias, 3 KB

    const int tid = threadIdx.x;
    const int tokBase = blockIdx.x * TOKS_PER_BLOCK;

    // Stage M (float4-coalesced) and bias into LDS
    {
        const float4* src = (const float4*)M;
        float4* dst = (float4*)&mLDS[0][0];
        #pragma unroll
        for (int i = tid; i < (16 * E_) / 4; i += 256) dst[i] = src[i];
        for (int i = tid; i < E_; i += 256) bLDS[i] = beff[i];
    }

    // Compute per-token features: 8 threads per token read x[token, 0..7]
    {
        int w = tid & 7;
        int t0 = tid >> 3;                       // 0..31
        for (int t = t0; t < TOKS_PER_BLOCK; t += 32) {
            float xv = x[(size_t)(tokBase + t) * E_ + w];
            float s, c;
            sincosf(xv, &s, &c);
            fLDS[t][w] = c;
            fLDS[t][8 + w] = s;
        }
    }
    __syncthreads();

    const int wave = tid >> 5;        // 0..7
    const int lane = tid & 31;
    const int row  = lane & 15;       // M-row (A) / N-col (B,C)
    const int half = lane >> 4;       // 0 or 1
    const int wtok = wave * 16;

    // A fragments: 16x4 f32 per k-block; lane<16 holds K=4kb..4kb+1,
    // lane>=16 holds K=4kb+2..4kb+3 (ISA 7.12.2 layout). Fixed for all tiles.
    v2f a[4];
    #pragma unroll
    for (int kb = 0; kb < 4; ++kb) {
        int cb = 4 * kb + 2 * half;
        a[kb].x = fLDS[wtok + row][cb];
        a[kb].y = fLDS[wtok + row][cb + 1];
    }

    float* outBase = out + (size_t)(tokBase + wtok) * E_;

    for (int ct = 0; ct < E_ / 16; ++ct) {
        const int col = ct * 16 + row;
        const float bias = bLDS[col];          // bias depends only on column
        v8f c;
        #pragma unroll
        for (int v = 0; v < 8; ++v) c[v] = bias;

        #pragma unroll
        for (int kb = 0; kb < 4; ++kb) {
            const int rb = 4 * kb + 2 * half;  // B: 4x16, mirrored layout
            v2f b;
            b.x = mLDS[rb][col];
            b.y = mLDS[rb + 1][col];
            c = __builtin_amdgcn_wmma_f32_16x16x4_f32(
                    false, a[kb], false, b, (short)0, c, false, false);
        }

        // C/D layout: VGPR v, lane<16 -> (m=v, n=row), lane>=16 -> (m=v+8, n=row)
        #pragma unroll
        for (int v = 0; v < 8; ++v)
            outBase[(size_t)(v + 8 * half) * E_ + col] = c[v];
    }
}

extern "C" void kernel_launch(void* const* d_in, const int* in_sizes, int n_in,
                              void* d_out, int out_size, void* d_ws, size_t ws_size,
                              hipStream_t stream) {
    const float* x     = (const float*)d_in[0];
    const float* theta = (const float*)d_in[1];
    const float* Wp    = (const float*)d_in[2];
    const float* bp    = (const float*)d_in[3];
    const float* Wo    = (const float*)d_in[4];
    const float* bo    = (const float*)d_in[5];
    float* out  = (float*)d_out;
    float* M    = (float*)d_ws;              // 16*768 floats
    float* beff = M + 16 * E_;               // 768 floats (52 KB total)

    qmha_prep<<<(17 * E_) / 256, 256, 0, stream>>>(theta, Wp, bp, Wo, bo, M, beff);
    qmha_main<<<NTOK / TOKS_PER_BLOCK, 256, 0, stream>>>(x, M, beff, out);
}